// RungeKuttaIntegratorCell_91336774516907
// MI455X (gfx1250) — compile-verified
//
#include <hip/hip_runtime.h>

// CDNA5 / gfx1250. wave32. f32 WMMA 16x16x4.
typedef __attribute__((ext_vector_type(2))) float v2f;
typedef __attribute__((ext_vector_type(4))) float v4f;
typedef __attribute__((ext_vector_type(8))) float v8f;

#define DT      0.002f
#define TSTEPS  4000
#define CHUNK   16
#define NCHUNK  (TSTEPS / CHUNK)     // 250 chunks total
#define NSEG    10                   // T split into 10 segments of 400 steps
#define SEGCH   (NCHUNK / NSEG)      // 25 chunks per segment
#define WROWS   48                   // 3 M-tiles of 16 (rows 34..39 and 42..47 zero pad)
#define WCOLS   36                   // K = 4 (state) + 32 (u) = 9 k-tiles of 4

__device__ __forceinline__ void mm4(const float A[4][4], const float B[4][4], float O[4][4]) {
    for (int i = 0; i < 4; ++i)
        for (int j = 0; j < 4; ++j) {
            float s = 0.f;
            for (int k = 0; k < 4; ++k) s += A[i][k] * B[k][j];
            O[i][j] = s;
        }
}

// ---------------------------------------------------------------------------
// Setup (1 thread): build the 48x36 chunk matrix W and Phi^(400) from C, M, K.
//   per-batch chunk map:  [y_0..y_15 ; x_end] = W * [x_start ; u_0..u_15]
//   x_{t+1} = Phi x_t + Psi u_t  (exact RK4 step of the LTI system)
// x_end rows placed at 32,33,40,41 so the D-tile2 fragment IS the next chunk's
// B-state fragment (zero-shuffle state carry).
// ---------------------------------------------------------------------------
__global__ void rk4_setup_W(const float* __restrict__ Cc, const float* __restrict__ Mm,
                            const float* __restrict__ Kk, float* __restrict__ W,
                            float* __restrict__ Wp /*Phi^(CHUNK*SEGCH), 16 floats*/) {
    if (threadIdx.x != 0 || blockIdx.x != 0) return;

    const float m0 = Mm[0], m1 = Mm[1];
    float F[4][4] = {};
    F[0][1] = 1.f;
    F[1][0] = -(Kk[0] + Kk[1]) / m0;
    F[1][1] = -(Cc[0] + Cc[1]) / m0;
    F[1][2] =  Kk[1] / m0;
    F[1][3] =  Cc[1] / m0;
    F[2][3] = 1.f;
    F[3][0] =  Kk[1] / m1;
    F[3][1] =  Cc[1] / m1;
    F[3][2] = -(Kk[2] + Kk[1]) / m1;
    F[3][3] = -(Cc[2] + Cc[1]) / m1;

    float hF[4][4], hF2[4][4], hF3[4][4], hF4[4][4];
    for (int i = 0; i < 4; ++i)
        for (int j = 0; j < 4; ++j) hF[i][j] = DT * F[i][j];
    mm4(hF, hF, hF2);
    mm4(hF2, hF, hF3);
    mm4(hF3, hF, hF4);

    float Phi[4][4], S[4][4], Psi[4][2];
    for (int i = 0; i < 4; ++i)
        for (int j = 0; j < 4; ++j) {
            const float id = (i == j) ? 1.f : 0.f;
            Phi[i][j] = id + hF[i][j] + 0.5f * hF2[i][j]
                      + (1.f / 6.f) * hF3[i][j] + (1.f / 24.f) * hF4[i][j];
            S[i][j] = DT * (id + 0.5f * hF[i][j]
                      + (1.f / 6.f) * hF2[i][j] + (1.f / 24.f) * hF3[i][j]);
        }
    for (int r = 0; r < 4; ++r) { Psi[r][0] = S[r][1] / m0; Psi[r][1] = S[r][3] / m1; }

    float P[CHUNK + 1][4][4];
    float Q[CHUNK][4][2];
    for (int i = 0; i < 4; ++i)
        for (int j = 0; j < 4; ++j) P[0][i][j] = (i == j) ? 1.f : 0.f;
    for (int p = 1; p <= CHUNK; ++p) mm4(Phi, P[p - 1], P[p]);
    for (int p = 0; p < CHUNK; ++p)
        for (int r = 0; r < 4; ++r)
            for (int c = 0; c < 2; ++c) {
                float s = 0.f;
                for (int k = 0; k < 4; ++k) s += P[p][r][k] * Psi[k][c];
                Q[p][r][c] = s;
            }

    for (int i = 0; i < WROWS * WCOLS; ++i) W[i] = 0.f;

    for (int j = 0; j < CHUNK; ++j)
        for (int c = 0; c < 2; ++c) {
            const int row = 2 * j + c;
            const int rc  = c ? 2 : 0;
            for (int k = 0; k < 4; ++k) W[row * WCOLS + k] = P[j + 1][rc][k];
            for (int i = 0; i <= j; ++i)
                for (int d = 0; d < 2; ++d)
                    W[row * WCOLS + 4 + 2 * i + d] = Q[j - i][rc][d];
        }
    for (int r = 0; r < 4; ++r) {
        const int row = 32 + (r & 1) + ((r >> 1) * 8);
        for (int k = 0; k < 4; ++k) W[row * WCOLS + k] = P[CHUNK][r][k];
        for (int i = 0; i < CHUNK; ++i)
            for (int d = 0; d < 2; ++d)
                W[row * WCOLS + 4 + 2 * i + d] = Q[CHUNK - 1 - i][r][d];
    }

    // Phi^(segment length) = P[CHUNK]^SEGCH for the cross-segment combine.
    float R[4][4], Tt[4][4];
    for (int i = 0; i < 4; ++i)
        for (int j = 0; j < 4; ++j) R[i][j] = (i == j) ? 1.f : 0.f;
    for (int s = 0; s < SEGCH; ++s) {
        mm4(P[CHUNK], R, Tt);
        for (int i = 0; i < 4; ++i)
            for (int j = 0; j < 4; ++j) R[i][j] = Tt[i][j];
    }
    for (int i = 0; i < 4; ++i)
        for (int j = 0; j < 4; ++j) Wp[i * 4 + j] = R[i][j];
}

// ---- chunk bodies (inlined; 8 affine b64 loads done by caller) -------------
__device__ __forceinline__ void chunk_state(const v2f A2[9], const v2f Bu[8], v2f& bstate) {
    v8f D2 = {};
    D2 = __builtin_amdgcn_wmma_f32_16x16x4_f32(false, A2[0], false, bstate, (short)0, D2, false, false);
#pragma unroll
    for (int t = 0; t < 8; ++t)
        D2 = __builtin_amdgcn_wmma_f32_16x16x4_f32(false, A2[t + 1], false, Bu[t], (short)0, D2, false, false);
    bstate[0] = D2[0];
    bstate[1] = D2[1];
}

__device__ __forceinline__ void chunk_full(const v2f A[3][9], const v2f Bu[8], v2f& bstate,
                                           float* __restrict__ q /*per-lane out ptr*/) {
    v8f D0 = {}; v8f D1 = {}; v8f D2 = {};
    D0 = __builtin_amdgcn_wmma_f32_16x16x4_f32(false, A[0][0], false, bstate, (short)0, D0, false, false);
    D1 = __builtin_amdgcn_wmma_f32_16x16x4_f32(false, A[1][0], false, bstate, (short)0, D1, false, false);
    D2 = __builtin_amdgcn_wmma_f32_16x16x4_f32(false, A[2][0], false, bstate, (short)0, D2, false, false);
#pragma unroll
    for (int t = 0; t < 8; ++t) {
        D0 = __builtin_amdgcn_wmma_f32_16x16x4_f32(false, A[0][t + 1], false, Bu[t], (short)0, D0, false, false);
        D1 = __builtin_amdgcn_wmma_f32_16x16x4_f32(false, A[1][t + 1], false, Bu[t], (short)0, D1, false, false);
        D2 = __builtin_amdgcn_wmma_f32_16x16x4_f32(false, A[2][t + 1], false, Bu[t], (short)0, D2, false, false);
    }
    bstate[0] = D2[0];
    bstate[1] = D2[1];
    // D0 = first 8 steps of the chunk (lane half picks 0-3 / 4-7), D1 = next 8.
    // Rows are (z1,z2) pairs in step order -> 8 contiguous floats per tile half.
    v4f s0 = { D0[0], D0[1], D0[2], D0[3] };
    v4f s1 = { D0[4], D0[5], D0[6], D0[7] };
    v4f s2 = { D1[0], D1[1], D1[2], D1[3] };
    v4f s3 = { D1[4], D1[5], D1[6], D1[7] };
    __builtin_nontemporal_store(s0, (v4f*)q);
    __builtin_nontemporal_store(s1, (v4f*)(q + 4));
    __builtin_nontemporal_store(s2, (v4f*)(q + 16));
    __builtin_nontemporal_store(s3, (v4f*)(q + 20));
}

// ---------------------------------------------------------------------------
// Pass 1: per (batch-group, segment) contribution c_s = x_end with x_start = 0.
// Only the state tile (9 WMMA / chunk). Segments 0..NSEG-2.
// Also warms L2 with u (64 MB < 192 MB L2) for pass 3.
// ---------------------------------------------------------------------------
__global__ void rk4_pass1(const float* __restrict__ u, const float* __restrict__ W,
                          float* __restrict__ c, int B) {
    const int lane = threadIdx.x & 31;
    const int wave = (blockIdx.x * blockDim.x + threadIdx.x) >> 5;
    const int nbg  = B >> 4;
    const int bg   = wave % nbg;
    const int seg  = wave / nbg;          // 0..NSEG-2
    const int half = lane >> 4;
    const int bcol = lane & 15;
    const int b    = bg * 16 + bcol;

    v2f A2[9];
#pragma unroll
    for (int kt = 0; kt < 9; ++kt)
        A2[kt] = *(const v2f*)(W + (32 + bcol) * WCOLS + kt * 4 + 2 * half);

    v2f bstate = { 0.f, 0.f };
    // per-lane u pointer for this segment; 8 loads at immediate offsets t*16B,
    // advanced by 128B per chunk (affine -> strength-reduced addressing)
    const float* p = u + (size_t)b * (TSTEPS * 2) + (size_t)(seg * SEGCH * CHUNK) * 2 + 2 * half;

    v2f Bu[8];
#pragma unroll
    for (int t = 0; t < 8; ++t) Bu[t] = *(const v2f*)(p + 4 * t);

    for (int ci = 0; ci < SEGCH - 1; ++ci) {
        v2f Bn[8];
#pragma unroll
        for (int t = 0; t < 8; ++t) Bn[t] = *(const v2f*)(p + 32 + 4 * t);  // next chunk
        chunk_state(A2, Bu, bstate);
#pragma unroll
        for (int t = 0; t < 8; ++t) Bu[t] = Bn[t];
        p += 32;
    }
    chunk_state(A2, Bu, bstate);   // peeled final chunk (no prefetch)

    // lane half h stores components (2h, 2h+1) of c_s for its batch
    *(v2f*)(c + ((size_t)seg * B + b) * 4 + 2 * half) = bstate;
}

// ---------------------------------------------------------------------------
// Pass 2: serial combine over segments (tiny): X_0 = x0; X_{s+1} = Phi^400 X_s + c_s.
// ---------------------------------------------------------------------------
__global__ void rk4_combine(const float* __restrict__ x0, const float* __restrict__ Wp,
                            const float* __restrict__ c, float* __restrict__ Xs, int B) {
    const int b = blockIdx.x * blockDim.x + threadIdx.x;
    if (b >= B) return;
    float P[16];
#pragma unroll
    for (int i = 0; i < 16; ++i) P[i] = Wp[i];
    float x[4];
#pragma unroll
    for (int r = 0; r < 4; ++r) { x[r] = x0[(size_t)b * 4 + r]; Xs[(size_t)b * 4 + r] = x[r]; }
    for (int s = 0; s < NSEG - 1; ++s) {
        float nx[4];
#pragma unroll
        for (int r = 0; r < 4; ++r) {
            float acc = c[((size_t)s * B + b) * 4 + r];
#pragma unroll
            for (int k = 0; k < 4; ++k) acc += P[r * 4 + k] * x[k];
            nx[r] = acc;
        }
#pragma unroll
        for (int r = 0; r < 4; ++r) {
            x[r] = nx[r];
            Xs[((size_t)(s + 1) * B + b) * 4 + r] = x[r];
        }
    }
}

// ---------------------------------------------------------------------------
// Pass 3: full chunked recurrence per (batch-group, segment) with known starts.
// 27 WMMA / chunk, double-buffered affine u loads, nontemporal b128 stores.
// ---------------------------------------------------------------------------
__global__ void rk4_pass3(const float* __restrict__ u, const float* __restrict__ Xs,
                          const float* __restrict__ W, float* __restrict__ out, int B) {
    const int lane = threadIdx.x & 31;
    const int wave = (blockIdx.x * blockDim.x + threadIdx.x) >> 5;
    const int nbg  = B >> 4;
    const int bg   = wave % nbg;
    const int seg  = wave / nbg;          // 0..NSEG-1
    const int half = lane >> 4;
    const int bcol = lane & 15;
    const int b    = bg * 16 + bcol;

    v2f A[3][9];
#pragma unroll
    for (int mt = 0; mt < 3; ++mt)
#pragma unroll
        for (int kt = 0; kt < 9; ++kt)
            A[mt][kt] = *(const v2f*)(W + (mt * 16 + bcol) * WCOLS + kt * 4 + 2 * half);

    // segment-start state in B-fragment layout: vgpr v, half h <- Xs[seg][b][2h+v]
    v2f bstate = *(const v2f*)(Xs + ((size_t)seg * B + b) * 4 + 2 * half);

    const size_t segoff = (size_t)(seg * SEGCH * CHUNK) * 2;
    const float* p = u   + (size_t)b * (TSTEPS * 2) + segoff + 2 * half;  // +128B per chunk
    float*       q = out + (size_t)b * (TSTEPS * 2) + segoff + 8 * half;  // +128B per chunk

    v2f Bu[8];
#pragma unroll
    for (int t = 0; t < 8; ++t) Bu[t] = *(const v2f*)(p + 4 * t);

    for (int ci = 0; ci < SEGCH - 1; ++ci) {
        v2f Bn[8];
#pragma unroll
        for (int t = 0; t < 8; ++t) Bn[t] = *(const v2f*)(p + 32 + 4 * t);  // next chunk
        chunk_full(A, Bu, bstate, q);
#pragma unroll
        for (int t = 0; t < 8; ++t) Bu[t] = Bn[t];
        p += 32;
        q += 32;
    }
    chunk_full(A, Bu, bstate, q);  // peeled final chunk (no prefetch)
}

extern "C" void kernel_launch(void* const* d_in, const int* in_sizes, int n_in,
                              void* d_out, int out_size, void* d_ws, size_t ws_size,
                              hipStream_t stream) {
    const float* u  = (const float*)d_in[0];
    const float* C  = (const float*)d_in[1];
    const float* M  = (const float*)d_in[2];
    const float* K  = (const float*)d_in[3];
    const float* x0 = (const float*)d_in[4];
    float* out = (float*)d_out;

    const int B   = in_sizes[4] / 4;   // 2048
    const int nbg = B / 16;            // 128 batch groups

    // workspace layout (floats)
    float* W  = (float*)d_ws;                      // 48*36
    float* Wp = W + WROWS * WCOLS;                 // 16 (Phi^400)
    float* c  = Wp + 16;                           // NSEG*B*4 segment contributions
    float* Xs = c + (size_t)NSEG * B * 4;          // NSEG*B*4 segment-start states

    rk4_setup_W<<<1, 32, 0, stream>>>(C, M, K, W, Wp);

    // Pass 1: (NSEG-1) segments x nbg groups, 1 wave each
    {
        const int waves = nbg * (NSEG - 1);        // 1152
        rk4_pass1<<<(waves * 32) / 256, 256, 0, stream>>>(u, W, c, B);
    }
    // Pass 2: serial segment combine, 1 thread per batch
    rk4_combine<<<(B + 255) / 256, 256, 0, stream>>>(x0, Wp, c, Xs, B);

    // Pass 3: NSEG segments x nbg groups, 1 wave each
    {
        const int waves = nbg * NSEG;              // 1280
        rk4_pass3<<<(waves * 32) / 256, 256, 0, stream>>>(u, Xs, W, out, B);
    }
}